// GCNEncoder_24481313587386
// MI455X (gfx1250) — compile-verified
//
#include <hip/hip_runtime.h>

// ---------------- device types ----------------
typedef __attribute__((ext_vector_type(8))) float v8f;
#if defined(__gfx1250__)
typedef __attribute__((ext_vector_type(2))) float v2f;
#endif

#define N_NODES 100000
#define N_EDGES 600000

// ---------------- utility kernels ----------------
__global__ void k_fill(float* __restrict__ p, long long n, float v) {
    long long i = (long long)blockIdx.x * blockDim.x + threadIdx.x;
    if (i < n) p[i] = v;
}

// deg[dst[e]] += 1  (deg pre-initialized to 1.0 for self-loop)
__global__ void k_deg(const long long* __restrict__ dst, float* __restrict__ deg, int E) {
    int e = blockIdx.x * blockDim.x + threadIdx.x;
    if (e < E) atomicAdd(&deg[(int)dst[e]], 1.0f);
}

__global__ void k_rsqrt_inplace(float* __restrict__ d, int n) {
    int i = blockIdx.x * blockDim.x + threadIdx.x;
    if (i < n) d[i] = rsqrtf(d[i]);
}

// ---------------- WMMA GEMM: C[M, NT*16] = A[M,K] * B[K, NT*16] ----------------
// 256-thread block = 8 waves. B staged once per block into LDS (b128 cooperative
// load). Each wave owns 16 rows and the FULL output width: one A fragment feeds
// NT back-to-back v_wmma_f32_16x16x4_f32 ops (A-reuse x NT).
template <int NT>
__global__ __launch_bounds__(256) void k_gemm_wide(const float* __restrict__ A,
                                                   const float* __restrict__ B,
                                                   float* __restrict__ C,
                                                   int M, int K) {
#if defined(__gfx1250__)
    constexpr int N = NT * 16;
    __shared__ float Bs[8192];           // up to K*N = 128*64 floats = 32 KB
    const int tid = threadIdx.x;

    // cooperative stage of the whole weight matrix into LDS
    const int total4 = (K * N) >> 2;
    const float4* B4 = (const float4*)B;
    float4* Bs4 = (float4*)Bs;
    for (int i = tid; i < total4; i += 256) Bs4[i] = B4[i];
    __syncthreads();

    const int lane = tid & 31;
    const int wave = tid >> 5;
    const int hi   = lane >> 4;          // 0: lanes 0-15, 1: lanes 16-31
    const int l16  = lane & 15;
    const int row0 = blockIdx.x * 128 + wave * 16;
    if (row0 >= M) return;               // wave-uniform; EXEC all-ones inside

    const float* arow = A + (size_t)(row0 + l16) * K;
    v8f acc[NT];
#pragma unroll
    for (int c = 0; c < NT; ++c) acc[c] = (v8f){};

    for (int k = 0; k < K; k += 4) {
        // A 16x4 f32 fragment: lanes 0-15 K=k,k+1 ; lanes 16-31 K=k+2,k+3
        const int ka = k + 2 * hi;
        v2f a;
        a.x = arow[ka];
        a.y = arow[ka + 1];
#pragma unroll
        for (int c = 0; c < NT; ++c) {
            v2f b;
            b.x = Bs[ka * N + c * 16 + l16];
            b.y = Bs[(ka + 1) * N + c * 16 + l16];
            acc[c] = __builtin_amdgcn_wmma_f32_16x16x4_f32(
                false, a, false, b, (short)0, acc[c], false, false);
        }
    }
    // C/D layout: VGPR r -> row = row0 + r + 8*hi, col = c*16 + l16
#pragma unroll
    for (int c = 0; c < NT; ++c)
#pragma unroll
        for (int r = 0; r < 8; ++r)
            C[(size_t)(row0 + r + 8 * hi) * N + c * 16 + l16] = acc[c][r];
#endif // __gfx1250__
}

// ---------------- edge-parallel normalized scatter-add ----------------
// VW features per thread (vectorized h read); per_edge = F/VW = 1<<pshift threads/edge.
// e in [0,E): real edges; e in [E,E+N): self-loops (src=dst=e-E).
template <int VW>
__global__ void k_scatter(const float* __restrict__ h,
                          const long long* __restrict__ src,
                          const long long* __restrict__ dst,
                          const float* __restrict__ dinv,
                          float* __restrict__ out,
                          int E, int N, int F, int pshift) {
    const long long gid = (long long)blockIdx.x * blockDim.x + threadIdx.x;
    const long long total = ((long long)(E + N)) << pshift;
    if (gid >= total) return;
    const int e  = (int)(gid >> pshift);
    const int fb = ((int)(gid & ((1 << pshift) - 1))) * VW;
    int s, d;
    if (e < E) { s = (int)src[e]; d = (int)dst[e]; }
    else       { s = d = e - E; }
    const float norm = dinv[s] * dinv[d];
    const float* hp = h + (size_t)s * F + fb;
    float* op = out + (size_t)d * F + fb;
    float hv[VW];
#pragma unroll
    for (int v = 0; v < VW; ++v) hv[v] = hp[v];       // coalesces to b64/b128
#pragma unroll
    for (int v = 0; v < VW; ++v) atomicAdd(&op[v], norm * hv[v]);
}

// x[i] = relu(x[i] + b[i & fmask])
__global__ void k_bias_relu(float* __restrict__ x, const float* __restrict__ b,
                            long long total, int fmask) {
    long long i = (long long)blockIdx.x * blockDim.x + threadIdx.x;
    if (i >= total) return;
    float v = x[i] + b[(int)(i & fmask)];
    x[i] = v > 0.0f ? v : 0.0f;
}

// Layer 3: H[n,0..1] = X[n,:32] @ W3[32,2] (2 cols can't fill a WMMA tile; ~1% of FLOPs)
__global__ void k_lin3(const float* __restrict__ X, const float* __restrict__ W,
                       float* __restrict__ H, int N) {
    int n = blockIdx.x * blockDim.x + threadIdx.x;
    if (n >= N) return;
    const float* xr = X + (size_t)n * 32;
    float a0 = 0.0f, a1 = 0.0f;
#pragma unroll
    for (int k = 0; k < 32; ++k) {
        float xv = xr[k];
        a0 = fmaf(xv, W[2 * k + 0], a0);
        a1 = fmaf(xv, W[2 * k + 1], a1);
    }
    H[(size_t)n * 2 + 0] = a0;
    H[(size_t)n * 2 + 1] = a1;
}

// ---------------- launch ----------------
extern "C" void kernel_launch(void* const* d_in, const int* in_sizes, int n_in,
                              void* d_out, int out_size, void* d_ws, size_t ws_size,
                              hipStream_t stream) {
    (void)in_sizes; (void)n_in; (void)out_size; (void)ws_size;
    const float*     x  = (const float*)d_in[0];
    const long long* ei = (const long long*)d_in[1];
    const float*     W1 = (const float*)d_in[2];
    const float*     b1 = (const float*)d_in[3];
    const float*     W2 = (const float*)d_in[4];
    const float*     b2 = (const float*)d_in[5];
    const float*     W3 = (const float*)d_in[6];
    const float*     b3 = (const float*)d_in[7];
    float*           out = (float*)d_out;

    const int N = N_NODES, E = N_EDGES;
    const long long* src = ei;
    const long long* dst = ei + E;

    // workspace: dinv | bufA (N*64: h1/h2/h3) | bufB (N*64: agg1/agg2)
    float* dinv = (float*)d_ws;
    float* bufA = dinv + N;
    float* bufB = bufA + (size_t)N * 64;

    const int T = 256;
    auto blk = [](long long n, int t) { return (unsigned)((n + t - 1) / t); };
    const unsigned gemmBlocks = (unsigned)((N + 127) / 128);   // 782

    // degrees (init 1.0 = self-loop), then dinv = rsqrt(deg) in place
    k_fill<<<blk(N, T), T, 0, stream>>>(dinv, N, 1.0f);
    k_deg<<<blk(E, T), T, 0, stream>>>(dst, dinv, E);
    k_rsqrt_inplace<<<blk(N, T), T, 0, stream>>>(dinv, N);

    // ---- layer 1: h1 = x @ W1 [128->64], aggregate, +b1, relu ----
    k_gemm_wide<4><<<gemmBlocks, 256, 0, stream>>>(x, W1, bufA, N, 128);
    k_fill<<<blk((long long)N * 64, T), T, 0, stream>>>(bufB, (long long)N * 64, 0.0f);
    k_scatter<4><<<blk(((long long)(E + N)) << 4, T), T, 0, stream>>>(
        bufA, src, dst, dinv, bufB, E, N, 64, 4);
    k_bias_relu<<<blk((long long)N * 64, T), T, 0, stream>>>(bufB, b1, (long long)N * 64, 63);

    // ---- layer 2: h2 = agg1 @ W2 [64->32], aggregate, +b2, relu ----
    k_gemm_wide<2><<<gemmBlocks, 256, 0, stream>>>(bufB, W2, bufA, N, 64);
    k_fill<<<blk((long long)N * 32, T), T, 0, stream>>>(bufB, (long long)N * 32, 0.0f);
    k_scatter<4><<<blk(((long long)(E + N)) << 3, T), T, 0, stream>>>(
        bufA, src, dst, dinv, bufB, E, N, 32, 3);
    k_bias_relu<<<blk((long long)N * 32, T), T, 0, stream>>>(bufB, b2, (long long)N * 32, 31);

    // ---- layer 3: h3 = agg2 @ W3 [32->2], aggregate into d_out, +b3, relu ----
    k_lin3<<<blk(N, T), T, 0, stream>>>(bufB, W3, bufA, N);
    k_fill<<<blk((long long)N * 2, T), T, 0, stream>>>(out, (long long)N * 2, 0.0f);
    k_scatter<2><<<blk((long long)(E + N), T), T, 0, stream>>>(
        bufA, src, dst, dinv, out, E, N, 2, 0);
    k_bias_relu<<<blk((long long)N * 2, T), T, 0, stream>>>(out, b3, (long long)N * 2, 1);
}